// CostConcatenation_61495341744707
// MI455X (gfx1250) — compile-verified
//
#include <hip/hip_runtime.h>
#include <cstdint>

typedef float v4f __attribute__((ext_vector_type(4)));
typedef int   v4i __attribute__((ext_vector_type(4)));

typedef __attribute__((address_space(1))) v4i* gas_v4i_ptr;
typedef __attribute__((address_space(3))) v4i* lds_v4i_ptr;

#define BB   2
#define HH   96
#define WW   192
#define CC   16
#define DD   128
#define C2   32          // 2*CC, contiguous innermost output dim
#define DTILE 8          // disparities per block
#define NTHREADS 256     // 8 waves (wave32)
#define WSTEP (NTHREADS / 8)          // 32 w-positions per sweep
#define NK   (WW / WSTEP)             // 6 sweeps cover the row
#define DSTRIDE_V4 ((size_t)HH * WW * C2 / 4)   // 147456 v4f between disparities

// Combined LDS image, element = v4f (16B), row = 4 elements (64B):
//   left  row w   -> offset w*4            (rows 0..191,   elems 0..767)
//   right row r   -> offset 768 + r*4      (rows 0..191,   elems 768..1535)
//   zero  row     -> offset 1536           (elems 1536..1539)
#define ROFF 768u
#define ZOFF 1536u
#define SELEMS 1540

#if defined(__has_builtin)
#if __has_builtin(__builtin_amdgcn_global_load_async_to_lds_b128)
#define HAVE_ASYNC_B128 1
#endif
#endif

__global__ __launch_bounds__(NTHREADS)
void cost_concat_kernel(const float* __restrict__ left,
                        const float* __restrict__ right,
                        float* __restrict__ out)
{
    __shared__ v4f S[SELEMS];        // 24.1 KB of the 320 KB/WGP LDS

    const int tid = threadIdx.x;
    int blk = blockIdx.x;
    const int dt = blk % (DD / DTILE);
    blk /= (DD / DTILE);
    const int h = blk % HH;
    const int b = blk / HH;

    // ---- Stage full left/right rows for (b,h) into LDS ----
    const v4f* gl = (const v4f*)(left  + (size_t)(b * HH + h) * WW * CC);
    const v4f* gr = (const v4f*)(right + (size_t)(b * HH + h) * WW * CC);

#if HAVE_ASYNC_B128
    // gfx1250 async DMA path: GLOBAL_LOAD_ASYNC_TO_LDS_B128 (ASYNCcnt).
#pragma unroll
    for (int i = tid; i < WW * CC / 4; i += NTHREADS) {
        __builtin_amdgcn_global_load_async_to_lds_b128(
            (gas_v4i_ptr)(uintptr_t)(gl + i),
            (lds_v4i_ptr)(uint32_t)(uintptr_t)&S[i],
            0, 0);
        __builtin_amdgcn_global_load_async_to_lds_b128(
            (gas_v4i_ptr)(uintptr_t)(gr + i),
            (lds_v4i_ptr)(uint32_t)(uintptr_t)&S[ROFF + i],
            0, 0);
    }
#else
#pragma unroll
    for (int i = tid; i < WW * CC / 4; i += NTHREADS) {
        S[i]        = gl[i];
        S[ROFF + i] = gr[i];
    }
#endif
    if (tid < 4) {                       // zero row for the invalid-redirect
        v4f z = {0.0f, 0.0f, 0.0f, 0.0f};
        S[ZOFF + tid] = z;
    }
#if HAVE_ASYNC_B128
    asm volatile("s_wait_asynccnt 0x0" ::: "memory");
#endif
    __syncthreads();

    // ---- Stream the output tile ----
    // out layout: ((((b*DD + d)*HH + h)*WW + w)*C2 + c)
    const int d0 = dt * DTILE;
    float* ob = out + ((size_t)(b * DD + d0) * HH + h) * WW * C2;

    // Fixed role per thread: quad 0..3 -> left half, 4..7 -> right half.
    const int  quad   = tid & 7;
    const int  cw0    = tid >> 3;                     // 0..31: starting w
    const bool isLeft = (quad < 4);
    const int  qq     = quad & 3;

    // Per-thread LDS element-offset base (v4f units); k adds 32*4 = 128 elems.
    //   left : row w   = cw0 + 32k          -> offL  + 128k
    //   right: row idx = cw0 + 112 - d + 32k-> offR(d)+ 128k
    uint32_t offbase = isLeft
        ? (uint32_t)(cw0 * 4 + qq)
        : (uint32_t)(ROFF + (uint32_t)(cw0 + 112 - d0) * 4 + qq);
    const uint32_t step = isLeft ? 0u : 4u;           // offbase -= step per dl

    // Zero-row redirect targets, pre-compensated for the +128k constant fold.
    uint32_t zoffk[NK];
#pragma unroll
    for (int k = 0; k < NK; ++k) zoffk[k] = (ZOFF + (uint32_t)qq) - (uint32_t)(k * 128);

    // validity: idx = cw0 + 112 - d + 32k in [0, 192)
    int idxb = cw0 + 112 - d0;

    // Two store bases so every (dl,k) offset fits the signed 24-bit store imm.
    v4f* dA = (v4f*)ob + (size_t)cw0 * (C2 / 4) + quad;
    v4f* dB = dA + (size_t)4 * DSTRIDE_V4;

#pragma unroll
    for (int dl = 0; dl < DTILE; ++dl) {
        // Gather NK values (batched ds_loads, no branches, no data selects).
        v4f vals[NK];
#pragma unroll
        for (int k = 0; k < NK; ++k) {
            const uint32_t idxk = (uint32_t)(idxb + WSTEP * k);   // 1 add
            const uint32_t sel  = (idxk < (uint32_t)WW)           // 1 cmp
                                ? offbase : zoffk[k];             // 1 cndmask
            vals[k] = S[sel + (uint32_t)(k * 128)];               // ds_load_b128 off:2048k
        }

        v4f*      dp   = (dl < 4) ? dA : dB;                      // unroll-time
        const int dloc = (dl < 4) ? dl : (dl - 4);
#pragma unroll
        for (int k = 0; k < NK; ++k) {
            __builtin_nontemporal_store(                          // b128 th:NT, imm offset
                vals[k],
                dp + (size_t)dloc * DSTRIDE_V4 + (size_t)k * WSTEP * (C2 / 4));
        }

        idxb   -= 1;      // d increases by 1
        offbase -= step;  // right rows shift by one row (4 elems); left unchanged
    }
}

extern "C" void kernel_launch(void* const* d_in, const int* in_sizes, int n_in,
                              void* d_out, int out_size, void* d_ws, size_t ws_size,
                              hipStream_t stream)
{
    const float* left  = (const float*)d_in[0];
    const float* right = (const float*)d_in[1];
    float* out = (float*)d_out;

    const int grid = BB * HH * (DD / DTILE);       // 2*96*16 = 3072 blocks
    cost_concat_kernel<<<grid, NTHREADS, 0, stream>>>(left, right, out);
}